// GAT_16449724744305
// MI455X (gfx1250) — compile-verified
//
#include <hip/hip_runtime.h>
#include <hip/hip_bf16.h>
#include <math.h>

// ---------------------------------------------------------------------------
// GAT forward (3x GATConv, 4 heads, concat=False) for MI455X / gfx1250.
// Dense projections use V_WMMA_F32_16X16X4_F32 (full fp32 precision; the
// GEMMs are ~8 GFLOP total and never the bottleneck). The edge scatter is
// kept L2-resident (h = 41MB, accum = 10MB << 192MB L2).
// ---------------------------------------------------------------------------

#define N_NODES 20000
#define N_EDGES 320000
#define N_GRAPHS 64
#define IN_DIM 39
#define HID 128
#define HEADS 4
#define HFD (HEADS * HID)   // 512
#define NUM_CONV 3
#define PRED_HID 32
#define NEG_SLOPE 0.2f

typedef __attribute__((ext_vector_type(2))) float v2f;
typedef __attribute__((ext_vector_type(8))) float v8f;

// ----------------------------- utility ------------------------------------

__global__ void fill_f32(float* __restrict__ p, float v, int n) {
    int i = blockIdx.x * blockDim.x + threadIdx.x;
    if (i < n) p[i] = v;
}

__device__ __forceinline__ void atomic_max_f32(float* addr, float val) {
    // monotonic int-bits trick: works with -inf initialization
    if (val >= 0.0f)
        atomicMax((int*)addr, __float_as_int(val));
    else
        atomicMin((unsigned int*)addr, __float_as_uint(val));
}

// ----------------------- embedding: x = nf @ emb_W + b ---------------------

__global__ void embed_kernel(const float* __restrict__ nf,
                             const float* __restrict__ W,
                             const float* __restrict__ b,
                             float* __restrict__ x) {
    __shared__ float fs[IN_DIM];
    const int n = blockIdx.x;
    const int t = threadIdx.x;   // 128 threads = one output column each
    if (t < IN_DIM) fs[t] = nf[(size_t)n * IN_DIM + t];
    __syncthreads();
    float acc = b[t];
#pragma unroll
    for (int k = 0; k < IN_DIM; ++k) acc += fs[k] * W[k * HID + t];
    x[(size_t)n * HID + t] = acc;
}

// ------------------ h = x @ W  via WMMA (M=20000,K=128,N=512) --------------
// Block = 256 threads = 8 wave32; wave w owns the 16x16 C-tile at column
// blockIdx.x*128 + w*16, rows blockIdx.y*16. 20000/16 = 1250 exactly, so
// no edge masking: EXEC stays all-ones as WMMA requires.

__global__ void gat_gemm_wmma(const float* __restrict__ X,   // [N,128]
                              const float* __restrict__ W,   // [128,512]
                              float* __restrict__ H) {       // [N,512]
    __shared__ float xs[16][132];                 // padded stride: no bank conflicts
    const int m0 = blockIdx.y * 16;
    const int nb = blockIdx.x * 128;
    const int t  = threadIdx.x;

    // cooperative load of the 16x128 A tile (8 contiguous floats per thread)
    {
        const int idx = t * 8;
        const int row = idx >> 7;
        const int col = idx & 127;
        const float* src = X + (size_t)(m0 + row) * HID + col;
#pragma unroll
        for (int i = 0; i < 8; ++i) xs[row][col + i] = src[i];
    }
    __syncthreads();

    const int wave = t >> 5;
    const int lane = t & 31;
    const int half = lane >> 4;      // 0: lanes 0-15, 1: lanes 16-31
    const int l16  = lane & 15;
    const int ncol = nb + wave * 16 + l16;

    v8f c = {};
#pragma unroll 4
    for (int k = 0; k < HID; k += 4) {
        // A 16x4 (ISA 7.12.2): lanes 0-15 hold K={k,k+1}, lanes 16-31 K={k+2,k+3}
        v2f a;
        a[0] = xs[l16][k + 2 * half];
        a[1] = xs[l16][k + 2 * half + 1];
        // B 4x16: rows of K striped across lanes, one row per half-wave per VGPR
        v2f bfrag;
        bfrag[0] = W[(size_t)(k + half) * HFD + ncol];
        bfrag[1] = W[(size_t)(k + half + 2) * HFD + ncol];
        c = __builtin_amdgcn_wmma_f32_16x16x4_f32(
                /*neg_a=*/false, a, /*neg_b=*/false, bfrag,
                /*c_mod=*/(short)0, c, /*reuse_a=*/false, /*reuse_b=*/false);
    }

    // C layout: VGPR r -> row r (lanes 0-15) / row r+8 (lanes 16-31), col lane&15
    float* out = H + (size_t)(m0 + half * 8) * HFD + ncol;
#pragma unroll
    for (int r = 0; r < 8; ++r) out[(size_t)r * HFD] = c[r];
}

// --------- per-node attention scores a_s[n,h], a_d[n,h] (dot over 128) -----

__global__ void att_scores(const float* __restrict__ h,
                           const float* __restrict__ asrc,  // [4,128]
                           const float* __restrict__ adst,  // [4,128]
                           float* __restrict__ a_s,
                           float* __restrict__ a_d) {
    const int n    = blockIdx.x;
    const int head = threadIdx.x >> 5;   // 128 threads: 4 waves, one head each
    const int lane = threadIdx.x & 31;
    const float* hv = h + (size_t)n * HFD + head * HID;
    const float* ws = asrc + head * HID;
    const float* wd = adst + head * HID;
    float ss = 0.f, sd = 0.f;
#pragma unroll
    for (int i = lane; i < HID; i += 32) {
        float hx = hv[i];
        ss += hx * ws[i];
        sd += hx * wd[i];
    }
#pragma unroll
    for (int off = 16; off; off >>= 1) {
        ss += __shfl_down(ss, off, 32);
        sd += __shfl_down(sd, off, 32);
    }
    if (lane == 0) {
        a_s[n * HEADS + head] = ss;
        a_d[n * HEADS + head] = sd;
    }
}

// ------------- edge pass A: leaky-relu score + segment max -----------------

__global__ void edge_score(const int* __restrict__ ei, int E, int Etot,
                           const float* __restrict__ a_s,
                           const float* __restrict__ a_d,
                           float* __restrict__ esc,
                           float* __restrict__ nmax) {
    int e = blockIdx.x * blockDim.x + threadIdx.x;
    if (e >= Etot) return;
    int s, d;
    if (e < E) { s = ei[e]; d = ei[E + e]; } else { s = d = e - E; }
#pragma unroll
    for (int hh = 0; hh < HEADS; ++hh) {
        float v = a_s[s * HEADS + hh] + a_d[d * HEADS + hh];
        v = v > 0.f ? v : NEG_SLOPE * v;
        esc[(size_t)e * HEADS + hh] = v;
        atomic_max_f32(&nmax[d * HEADS + hh], v);
    }
}

// ------------- edge pass B: exp(e - max) + segment sum ---------------------

__global__ void edge_exp(const int* __restrict__ ei, int E, int Etot,
                         const float* __restrict__ nmax,
                         float* __restrict__ esc,      // in: score, out: weight
                         float* __restrict__ denom) {
    int e = blockIdx.x * blockDim.x + threadIdx.x;
    if (e >= Etot) return;
    int d;
    if (e < E) { d = ei[E + e]; } else { d = e - E; }
#pragma unroll
    for (int hh = 0; hh < HEADS; ++hh) {
        float w = expf(esc[(size_t)e * HEADS + hh] - nmax[d * HEADS + hh]);
        esc[(size_t)e * HEADS + hh] = w;
        atomicAdd(&denom[d * HEADS + hh], w);
    }
}

// ---- edge pass C: out[dst,f] += sum_h alpha_h * h[src,h,f] (head-fused) ----

__global__ void edge_scatter(const int* __restrict__ ei, int E, int Etot,
                             const float* __restrict__ h,
                             const float* __restrict__ w,
                             const float* __restrict__ denom,
                             float* __restrict__ xn) {
    __shared__ float alpha[HEADS];
    for (int e = blockIdx.x; e < Etot; e += gridDim.x) {
        int s, d;
        if (e < E) { s = ei[e]; d = ei[E + e]; } else { s = d = e - E; }
        if (threadIdx.x < HEADS) {
            alpha[threadIdx.x] =
                w[(size_t)e * HEADS + threadIdx.x] / denom[d * HEADS + threadIdx.x];
        }
        __syncthreads();
        const int f = threadIdx.x;   // 128 threads
        const float* hs = h + (size_t)s * HFD;
        float v = alpha[0] * hs[f] + alpha[1] * hs[HID + f] +
                  alpha[2] * hs[2 * HID + f] + alpha[3] * hs[3 * HID + f];
        atomicAdd(&xn[(size_t)d * HID + f], v);
        __syncthreads();
    }
}

// ------------- finalize layer: mean over heads + bias ----------------------

__global__ void layer_finalize(const float* __restrict__ xn,
                               const float* __restrict__ bias,  // [128]
                               float* __restrict__ x, int n) {
    int i = blockIdx.x * blockDim.x + threadIdx.x;
    if (i >= n) return;
    x[i] = xn[i] * (1.0f / HEADS) + bias[i & (HID - 1)];
}

// ------------- global mean pool (scatter) ----------------------------------

__global__ void pool_scatter(const float* __restrict__ x,
                             const int* __restrict__ batch,
                             float* __restrict__ gsum,
                             float* __restrict__ gcnt, int n) {
    int i = blockIdx.x * blockDim.x + threadIdx.x;
    if (i >= n) return;
    int node = i >> 7;          // /128
    int f    = i & (HID - 1);
    int g    = batch[node];
    atomicAdd(&gsum[(size_t)g * HID + f], x[i]);
    if (f == 0) atomicAdd(&gcnt[g], 1.0f);
}

// ------------- head: (gsum/cnt) @ fc_W + fc_b, then @ out_W + out_b --------

__global__ void head_mlp(const float* __restrict__ gsum,
                         const float* __restrict__ gcnt,
                         const float* __restrict__ fcW,   // [128,32]
                         const float* __restrict__ fcb,   // [32]
                         const float* __restrict__ outW,  // [32,1]
                         const float* __restrict__ outb,  // [1]
                         float* __restrict__ out) {
    const int g = blockIdx.x;       // 64 graphs
    const int t = threadIdx.x;      // 32 threads = PRED_HID
    float cnt = gcnt[g];
    float inv = 1.0f / (cnt > 1.0f ? cnt : 1.0f);
    float acc = fcb[t];
#pragma unroll 4
    for (int k = 0; k < HID; ++k)
        acc += gsum[(size_t)g * HID + k] * inv * fcW[k * PRED_HID + t];
    float v = acc * outW[t];
#pragma unroll
    for (int off = 16; off; off >>= 1) v += __shfl_down(v, off, 32);
    if (t == 0) out[g] = v + outb[0];
}

// ---------------------------------------------------------------------------

extern "C" void kernel_launch(void* const* d_in, const int* in_sizes, int n_in,
                              void* d_out, int out_size, void* d_ws, size_t ws_size,
                              hipStream_t stream) {
    const float* node_feat = (const float*)d_in[0];
    const int*   edge_idx  = (const int*)d_in[1];
    const int*   batch     = (const int*)d_in[2];
    const float* emb_W     = (const float*)d_in[3];
    const float* emb_b     = (const float*)d_in[4];
    const float* lin_W     = (const float*)d_in[5];   // [3,128,512]
    const float* att_src   = (const float*)d_in[6];   // [3,4,128]
    const float* att_dst   = (const float*)d_in[7];   // [3,4,128]
    const float* conv_b    = (const float*)d_in[8];   // [3,128]
    const float* fc_W      = (const float*)d_in[9];
    const float* fc_b      = (const float*)d_in[10];
    const float* out_W     = (const float*)d_in[11];
    const float* out_b     = (const float*)d_in[12];
    float* out = (float*)d_out;

    const int E    = in_sizes[1] / 2;       // 320000
    const int Etot = E + N_NODES;           // + self loops

    // ---- workspace carving (floats) ----
    float* ws = (float*)d_ws;
    float* x     = ws;                                   ws += (size_t)N_NODES * HID;   // 2.56M
    float* h     = ws;                                   ws += (size_t)N_NODES * HFD;   // 10.24M
    float* a_s   = ws;                                   ws += (size_t)N_NODES * HEADS;
    float* a_d   = ws;                                   ws += (size_t)N_NODES * HEADS;
    float* esc   = ws;                                   ws += (size_t)Etot * HEADS;
    float* nmax  = ws;                                   ws += (size_t)N_NODES * HEADS;
    float* denom = ws;                                   ws += (size_t)N_NODES * HEADS;
    float* xn    = ws;                                   ws += (size_t)N_NODES * HID;
    float* gsum  = ws;                                   ws += (size_t)N_GRAPHS * HID;
    float* gcnt  = ws;                                   ws += N_GRAPHS;

    const int NH  = N_NODES * HEADS;        // 80000
    const int NF  = N_NODES * HID;          // 2,560,000
    const int EB  = (Etot + 255) / 256;

    // ---- embedding ----
    embed_kernel<<<N_NODES, HID, 0, stream>>>(node_feat, emb_W, emb_b, x);

    // ---- GAT conv layers ----
    for (int l = 0; l < NUM_CONV; ++l) {
        const float* Wl  = lin_W  + (size_t)l * HID * HFD;
        const float* asl = att_src + (size_t)l * HEADS * HID;
        const float* adl = att_dst + (size_t)l * HEADS * HID;
        const float* bl  = conv_b + (size_t)l * HID;

        // h = x @ Wl  (WMMA fp32)
        gat_gemm_wmma<<<dim3(HFD / 128, N_NODES / 16), 256, 0, stream>>>(x, Wl, h);

        // per-node attention dot products
        att_scores<<<N_NODES, 128, 0, stream>>>(h, asl, adl, a_s, a_d);

        // init segment buffers
        fill_f32<<<(NH + 255) / 256, 256, 0, stream>>>(nmax, -INFINITY, NH);
        fill_f32<<<(NH + 255) / 256, 256, 0, stream>>>(denom, 0.0f, NH);
        fill_f32<<<(NF + 255) / 256, 256, 0, stream>>>(xn, 0.0f, NF);

        // segment softmax over incoming edges
        edge_score<<<EB, 256, 0, stream>>>(edge_idx, E, Etot, a_s, a_d, esc, nmax);
        edge_exp  <<<EB, 256, 0, stream>>>(edge_idx, E, Etot, nmax, esc, denom);

        // message scatter (head-fused: 128 atomics per edge)
        edge_scatter<<<Etot, HID, 0, stream>>>(edge_idx, E, Etot, h, esc, denom, xn);

        // head mean + bias -> next x
        layer_finalize<<<(NF + 255) / 256, 256, 0, stream>>>(xn, bl, x, NF);
    }

    // ---- global mean pool + head MLP ----
    fill_f32<<<(N_GRAPHS * HID + 255) / 256, 256, 0, stream>>>(gsum, 0.0f, N_GRAPHS * HID);
    fill_f32<<<1, 64, 0, stream>>>(gcnt, 0.0f, N_GRAPHS);
    pool_scatter<<<(NF + 255) / 256, 256, 0, stream>>>(x, batch, gsum, gcnt, NF);
    head_mlp<<<N_GRAPHS, PRED_HID, 0, stream>>>(gsum, gcnt, fc_W, fc_b, out_W, out_b, out);
}